// YourGNNModel_83365315215963
// MI455X (gfx1250) — compile-verified
//
#include <hip/hip_runtime.h>
#include <math.h>

#define IN_DIM    128
#define HID       32
#define HEADS     4
#define OUT_DIM   40
#define NEG_SLOPE 0.2f

// LDS row stride for W1 tile: need 2*S % 64 == 16 so the two half-wave groups
// (rows kb and kb+2) land on disjoint LDS bank sets. 136 = 128 + 8 pad.
#define W1_LDS_STRIDE 136

typedef __attribute__((ext_vector_type(2))) float v2f;
typedef __attribute__((ext_vector_type(8))) float v8f;

__device__ __forceinline__ float lrelu(float x) {
  return x > 0.f ? x : NEG_SLOPE * x;
}

// Float atomic max via integer CAS (robust; only used in cheap pass-1 kernels).
__device__ __forceinline__ void atomicMaxF32(float* addr, float val) {
  int* ia = reinterpret_cast<int*>(addr);
  int old = __float_as_int(*addr);
  while (__int_as_float(old) < val) {
    int assumed = old;
    old = atomicCAS(ia, assumed, __float_as_int(val));
    if (old == assumed) break;
  }
}

// ---------------------------------------------------------------------------
// Generic fill
// ---------------------------------------------------------------------------
__global__ void fill_f32(float* __restrict__ p, float v, size_t n) {
  size_t i = (size_t)blockIdx.x * blockDim.x + threadIdx.x;
  size_t stride = (size_t)gridDim.x * blockDim.x;
  for (; i < n; i += stride) p[i] = v;
}

// ---------------------------------------------------------------------------
// GEMM1: feat1[N x 128] = H[N x 128] @ W1[128 x 128]   (fp32 WMMA 16x16x4)
// W1 staged in LDS once per block (bank-conflict-free padded stride).
// One wave -> 16-row strip x 128 cols (8 tiles of 16x16).
// A layout (16x4 f32): lanes 0-15 hold {K=0,K=1}, lanes 16-31 hold {K=2,K=3}.
// B layout (4x16 f32): vgpr0 = row K=0 (lanes 0-15) / K=2 (lanes 16-31), etc.
// ---------------------------------------------------------------------------
__global__ void __launch_bounds__(256)
gat_gemm1(const float* __restrict__ H, const float* __restrict__ W,
          float* __restrict__ outF, int N) {
  __shared__ float Ws[128 * W1_LDS_STRIDE];

  // Cooperative coalesced load of W1 (128x128 f32) into padded LDS.
  for (int i = threadIdx.x; i < (128 * 128) / 4; i += 256) {
    const int elem = i * 4;
    const int row = elem >> 7;       // /128
    const int col = elem & 127;
    const float4 v = *(const float4*)(W + elem);
    float* p = Ws + row * W1_LDS_STRIDE + col;
    p[0] = v.x; p[1] = v.y; p[2] = v.z; p[3] = v.w;
  }
  __syncthreads();

  const int lane = threadIdx.x & 31;
  const int wave = threadIdx.x >> 5;
  const int strip = blockIdx.x * 8 + wave;
  const int row0 = strip * 16;
  if (row0 >= N) return;                 // wave-uniform exit, EXEC stays full
  const int half = lane >> 4;
  const int idx  = lane & 15;
  const int r    = row0 + idx;

  v8f acc[8] = {};
  for (int k0 = 0; k0 < IN_DIM; k0 += 4) {
    const int kb = k0 + 2 * half;
    v2f a;
    if (r < N) {
      const float* hp = H + (size_t)r * IN_DIM + kb;
      a.x = hp[0]; a.y = hp[1];
    } else {
      a.x = 0.f; a.y = 0.f;
    }
    const float* w0 = Ws + kb * W1_LDS_STRIDE;
    const float* w1 = Ws + (kb + 1) * W1_LDS_STRIDE;
#pragma unroll
    for (int t = 0; t < 8; ++t) {
      const int col = t * 16 + idx;
      v2f b;
      b.x = w0[col];
      b.y = w1[col];
      acc[t] = __builtin_amdgcn_wmma_f32_16x16x4_f32(
          false, a, false, b, (short)0, acc[t], false, false);
    }
  }
#pragma unroll
  for (int j = 0; j < 8; ++j) {
    const int rr = row0 + half * 8 + j;
    if (rr < N) {
#pragma unroll
      for (int t = 0; t < 8; ++t)
        outF[(size_t)rr * IN_DIM + t * 16 + idx] = acc[t][j];
    }
  }
}

// ---------------------------------------------------------------------------
// GEMM2: feat2[N x 40] = H1[N x 128] @ W2[128 x 40]  (3 col tiles, 48 padded)
// W2 staged flat in LDS (natural stride 40 already gives 2*40%64==16 -> no
// half-wave bank overlap).
// ---------------------------------------------------------------------------
__global__ void __launch_bounds__(256)
gat_gemm2(const float* __restrict__ H, const float* __restrict__ W,
          float* __restrict__ outF, int N) {
  __shared__ float Ws[128 * OUT_DIM];

  for (int i = threadIdx.x; i < (128 * OUT_DIM) / 4; i += 256) {
    const int elem = i * 4;
    const float4 v = *(const float4*)(W + elem);
    float* p = Ws + elem;
    p[0] = v.x; p[1] = v.y; p[2] = v.z; p[3] = v.w;
  }
  __syncthreads();

  const int lane = threadIdx.x & 31;
  const int wave = threadIdx.x >> 5;
  const int strip = blockIdx.x * 8 + wave;
  const int row0 = strip * 16;
  if (row0 >= N) return;
  const int half = lane >> 4;
  const int idx  = lane & 15;
  const int r    = row0 + idx;

  v8f acc[3] = {};
  for (int k0 = 0; k0 < IN_DIM; k0 += 4) {
    const int kb = k0 + 2 * half;
    v2f a;
    if (r < N) {
      const float* hp = H + (size_t)r * IN_DIM + kb;
      a.x = hp[0]; a.y = hp[1];
    } else {
      a.x = 0.f; a.y = 0.f;
    }
    const float* w0 = Ws + kb * OUT_DIM;
    const float* w1 = Ws + (kb + 1) * OUT_DIM;
#pragma unroll
    for (int t = 0; t < 3; ++t) {
      const int col = t * 16 + idx;
      v2f b;
      b.x = (col < OUT_DIM) ? w0[col] : 0.f;
      b.y = (col < OUT_DIM) ? w1[col] : 0.f;
      acc[t] = __builtin_amdgcn_wmma_f32_16x16x4_f32(
          false, a, false, b, (short)0, acc[t], false, false);
    }
  }
#pragma unroll
  for (int j = 0; j < 8; ++j) {
    const int rr = row0 + half * 8 + j;
    if (rr < N) {
#pragma unroll
      for (int t = 0; t < 3; ++t) {
        const int col = t * 16 + idx;
        if (col < OUT_DIM) outF[(size_t)rr * OUT_DIM + col] = acc[t][j];
      }
    }
  }
}

// ---------------------------------------------------------------------------
// Attention scores, layer 1: el/er[N x 4] = per-head dot(feat, al/ar)
// ---------------------------------------------------------------------------
__global__ void attn_scores1(const float* __restrict__ feat,
                             const float* __restrict__ al,
                             const float* __restrict__ ar,
                             float* __restrict__ el, float* __restrict__ er,
                             int N) {
  int g = blockIdx.x * blockDim.x + threadIdx.x;  // n*HEADS + head
  if (g >= N * HEADS) return;
  int n = g >> 2, hd = g & 3;
  const float* f = feat + (size_t)n * IN_DIM + hd * HID;
  const float* a = al + hd * HID;
  const float* b = ar + hd * HID;
  float sl = 0.f, sr = 0.f;
#pragma unroll
  for (int d = 0; d < HID; ++d) { float v = f[d]; sl += v * a[d]; sr += v * b[d]; }
  el[g] = sl; er[g] = sr;
}

// Attention scores, layer 2 (1 head, OUT_DIM wide)
__global__ void attn_scores2(const float* __restrict__ feat,
                             const float* __restrict__ al,
                             const float* __restrict__ ar,
                             float* __restrict__ el, float* __restrict__ er,
                             int N) {
  int n = blockIdx.x * blockDim.x + threadIdx.x;
  if (n >= N) return;
  const float* f = feat + (size_t)n * OUT_DIM;
  float sl = 0.f, sr = 0.f;
#pragma unroll
  for (int d = 0; d < OUT_DIM; ++d) { float v = f[d]; sl += v * al[d]; sr += v * ar[d]; }
  el[n] = sl; er[n] = sr;
}

// ---------------------------------------------------------------------------
// Layer-1 edge softmax passes
// ---------------------------------------------------------------------------
__global__ void edge_max1(const int* __restrict__ src, const int* __restrict__ dst,
                          const float* __restrict__ el, const float* __restrict__ er,
                          float* __restrict__ eraw, float* __restrict__ emax, int E) {
  int e = blockIdx.x * blockDim.x + threadIdx.x;
  if (e >= E) return;
  int s = src[e], d = dst[e];
  float4 l = *(const float4*)(el + (size_t)s * HEADS);
  float4 r = *(const float4*)(er + (size_t)d * HEADS);
  float4 ev;
  ev.x = lrelu(l.x + r.x); ev.y = lrelu(l.y + r.y);
  ev.z = lrelu(l.z + r.z); ev.w = lrelu(l.w + r.w);
  *(float4*)(eraw + (size_t)e * HEADS) = ev;
  float* m = emax + (size_t)d * HEADS;
  atomicMaxF32(m + 0, ev.x); atomicMaxF32(m + 1, ev.y);
  atomicMaxF32(m + 2, ev.z); atomicMaxF32(m + 3, ev.w);
}

__global__ void fix_max(float* __restrict__ emax, size_t n) {
  size_t i = (size_t)blockIdx.x * blockDim.x + threadIdx.x;
  size_t stride = (size_t)gridDim.x * blockDim.x;
  for (; i < n; i += stride) {
    float v = emax[i];
    if (!isfinite(v)) emax[i] = 0.f;
  }
}

__global__ void edge_exp1(const int* __restrict__ dst, float* __restrict__ eraw,
                          const float* __restrict__ emax, float* __restrict__ den,
                          int E) {
  int e = blockIdx.x * blockDim.x + threadIdx.x;
  if (e >= E) return;
  int d = dst[e];
  float4 ev = *(float4*)(eraw + (size_t)e * HEADS);
  float4 m = *(const float4*)(emax + (size_t)d * HEADS);
  ev.x = expf(ev.x - m.x); ev.y = expf(ev.y - m.y);
  ev.z = expf(ev.z - m.z); ev.w = expf(ev.w - m.w);
  *(float4*)(eraw + (size_t)e * HEADS) = ev;
  float* dn = den + (size_t)d * HEADS;
  atomicAdd(dn + 0, ev.x); atomicAdd(dn + 1, ev.y);
  atomicAdd(dn + 2, ev.z); atomicAdd(dn + 3, ev.w);
}

// Layer-1 aggregate: one wave per edge, lane covers 4 features (head = lane/8).
__global__ void __launch_bounds__(256)
edge_aggr1(const int* __restrict__ src, const int* __restrict__ dst,
           const float* __restrict__ feat, const float* __restrict__ ee,
           const float* __restrict__ den, float* __restrict__ outH, int E) {
  const int lane = threadIdx.x & 31;
  const int e = blockIdx.x * 8 + (threadIdx.x >> 5);
  if (e >= E) return;
  const int s = src[e], d = dst[e];
  const int hd = lane >> 3;
  const float alpha = ee[(size_t)e * HEADS + hd] /
                      fmaxf(den[(size_t)d * HEADS + hd], 1e-9f);
  const float4 v = *(const float4*)(feat + (size_t)s * IN_DIM + lane * 4);
  float* o = outH + (size_t)d * IN_DIM + lane * 4;
  atomicAdd(o + 0, v.x * alpha);
  atomicAdd(o + 1, v.y * alpha);
  atomicAdd(o + 2, v.z * alpha);
  atomicAdd(o + 3, v.w * alpha);
}

// h1 = relu(agg + b1)
__global__ void bias_relu(float* __restrict__ h1, const float* __restrict__ b1,
                          size_t total) {
  size_t i = (size_t)blockIdx.x * blockDim.x + threadIdx.x;
  size_t stride = (size_t)gridDim.x * blockDim.x;
  for (; i < total; i += stride)
    h1[i] = fmaxf(h1[i] + b1[i & (IN_DIM - 1)], 0.f);
}

// ---------------------------------------------------------------------------
// Layer-2 edge softmax passes (1 head)
// ---------------------------------------------------------------------------
__global__ void edge_max2(const int* __restrict__ src, const int* __restrict__ dst,
                          const float* __restrict__ el, const float* __restrict__ er,
                          float* __restrict__ eraw, float* __restrict__ emax, int E) {
  int e = blockIdx.x * blockDim.x + threadIdx.x;
  if (e >= E) return;
  float ev = lrelu(el[src[e]] + er[dst[e]]);
  eraw[e] = ev;
  atomicMaxF32(emax + dst[e], ev);
}

__global__ void edge_exp2(const int* __restrict__ dst, float* __restrict__ eraw,
                          const float* __restrict__ emax, float* __restrict__ den,
                          int E) {
  int e = blockIdx.x * blockDim.x + threadIdx.x;
  if (e >= E) return;
  int d = dst[e];
  float ev = expf(eraw[e] - emax[d]);
  eraw[e] = ev;
  atomicAdd(den + d, ev);
}

// Layer-2 aggregate: one thread per (edge, feature-pair), OUT_DIM=40 -> 20 pairs.
__global__ void edge_aggr2(const int* __restrict__ src, const int* __restrict__ dst,
                           const float* __restrict__ feat, const float* __restrict__ ee,
                           const float* __restrict__ den, float* __restrict__ outO,
                           int E) {
  long long g = (long long)blockIdx.x * blockDim.x + threadIdx.x;
  if (g >= (long long)E * (OUT_DIM / 2)) return;
  int e = (int)(g / (OUT_DIM / 2));
  int j = (int)(g % (OUT_DIM / 2)) * 2;
  int s = src[e], d = dst[e];
  float alpha = ee[e] / fmaxf(den[d], 1e-9f);
  float2 v = *(const float2*)(feat + (size_t)s * OUT_DIM + j);
  float* o = outO + (size_t)d * OUT_DIM + j;
  atomicAdd(o + 0, v.x * alpha);
  atomicAdd(o + 1, v.y * alpha);
}

__global__ void add_bias_out(float* __restrict__ outO, const float* __restrict__ b2,
                             size_t total) {
  size_t i = (size_t)blockIdx.x * blockDim.x + threadIdx.x;
  size_t stride = (size_t)gridDim.x * blockDim.x;
  for (; i < total; i += stride) outO[i] += b2[i % OUT_DIM];
}

// ---------------------------------------------------------------------------
// Launch
// ---------------------------------------------------------------------------
extern "C" void kernel_launch(void* const* d_in, const int* in_sizes, int n_in,
                              void* d_out, int out_size, void* d_ws, size_t ws_size,
                              hipStream_t stream) {
  (void)n_in; (void)out_size; (void)ws_size;
  const float* h   = (const float*)d_in[0];
  const int*   src = (const int*)d_in[1];
  const int*   dst = (const int*)d_in[2];
  const float* W1  = (const float*)d_in[3];
  const float* al1 = (const float*)d_in[4];
  const float* ar1 = (const float*)d_in[5];
  const float* b1  = (const float*)d_in[6];
  const float* W2  = (const float*)d_in[7];
  const float* al2 = (const float*)d_in[8];
  const float* ar2 = (const float*)d_in[9];
  const float* b2  = (const float*)d_in[10];
  float* out = (float*)d_out;
  const int N = in_sizes[0] / IN_DIM;
  const int E = in_sizes[1];

  // Workspace layout (layer-2 buffers alias layer-1 buffers; smaller or equal).
  char* ws = (char*)d_ws;
  size_t off = 0;
  auto take = [&](size_t bytes) {
    char* p = ws + off;
    off = (off + bytes + 255) & ~(size_t)255;
    return p;
  };
  float* feat1 = (float*)take((size_t)N * IN_DIM * sizeof(float));
  float* h1    = (float*)take((size_t)N * IN_DIM * sizeof(float));
  float* el1   = (float*)take((size_t)N * HEADS * sizeof(float));
  float* er1   = (float*)take((size_t)N * HEADS * sizeof(float));
  float* emax1 = (float*)take((size_t)N * HEADS * sizeof(float));
  float* den1  = (float*)take((size_t)N * HEADS * sizeof(float));
  float* eraw1 = (float*)take((size_t)E * HEADS * sizeof(float));
  // Aliases for layer 2 (used strictly after layer 1 completes on the stream).
  float* feat2 = feat1;   // N*40 <= N*128
  float* el2   = el1;
  float* er2   = er1;
  float* emax2 = emax1;
  float* den2  = den1;
  float* eraw2 = eraw1;

  const float NEG_INF = -__builtin_inff();
  const int edgeBlocks = (E + 255) / 256;
  const int strips = (N + 15) / 16;
  const int gemmBlocks = (strips + 7) / 8;

  // ---- init ----
  fill_f32<<<1024, 256, 0, stream>>>(h1, 0.f, (size_t)N * IN_DIM);
  fill_f32<<<1024, 256, 0, stream>>>(emax1, NEG_INF, (size_t)N * HEADS);
  fill_f32<<<1024, 256, 0, stream>>>(den1, 0.f, (size_t)N * HEADS);
  fill_f32<<<1024, 256, 0, stream>>>(out, 0.f, (size_t)N * OUT_DIM);

  // ---- layer 1 ----
  gat_gemm1<<<gemmBlocks, 256, 0, stream>>>(h, W1, feat1, N);
  attn_scores1<<<(N * HEADS + 255) / 256, 256, 0, stream>>>(feat1, al1, ar1, el1, er1, N);
  edge_max1<<<edgeBlocks, 256, 0, stream>>>(src, dst, el1, er1, eraw1, emax1, E);
  fix_max<<<1024, 256, 0, stream>>>(emax1, (size_t)N * HEADS);
  edge_exp1<<<edgeBlocks, 256, 0, stream>>>(dst, eraw1, emax1, den1, E);
  edge_aggr1<<<(E + 7) / 8, 256, 0, stream>>>(src, dst, feat1, eraw1, den1, h1, E);
  bias_relu<<<1024, 256, 0, stream>>>(h1, b1, (size_t)N * IN_DIM);

  // ---- layer 2 ----
  gat_gemm2<<<gemmBlocks, 256, 0, stream>>>(h1, W2, feat2, N);
  attn_scores2<<<(N + 255) / 256, 256, 0, stream>>>(feat2, al2, ar2, el2, er2, N);
  fill_f32<<<1024, 256, 0, stream>>>(emax2, NEG_INF, (size_t)N);
  fill_f32<<<1024, 256, 0, stream>>>(den2, 0.f, (size_t)N);
  edge_max2<<<edgeBlocks, 256, 0, stream>>>(src, dst, el2, er2, eraw2, emax2, E);
  fix_max<<<1024, 256, 0, stream>>>(emax2, (size_t)N);
  edge_exp2<<<edgeBlocks, 256, 0, stream>>>(dst, eraw2, emax2, den2, E);
  {
    long long total = (long long)E * (OUT_DIM / 2);
    int blocks = (int)((total + 255) / 256);
    edge_aggr2<<<blocks, 256, 0, stream>>>(src, dst, feat2, eraw2, den2, out, E);
  }
  add_bias_out<<<1024, 256, 0, stream>>>(out, b2, (size_t)N * OUT_DIM);
}